// CrossNetwork_54365696033285
// MI455X (gfx1250) — compile-verified
//
#include <hip/hip_runtime.h>

// ---------------------------------------------------------------------------
// CrossNetwork (DCN) on MI455X / gfx1250.
//
// Math: x_{i+1} = x0 * dot(x_i, W_i) + b_i + x_i  collapses to
//   x_i = a_i * x0 + c_i,  c_i = sum_{j<i} b_j  (row-independent)
//   s_i = a_i * d_i + e_i, d_i = dot(x0_row, W_i), e_i = dot(c_i, W_i)
//   a_{i+1} = a_i + s_i, a_0 = 1;  out = a_4 * x0 + csum
// => one pass over x for d (WMMA f32 16x16x4 GEMM), one pass for the output.
// ---------------------------------------------------------------------------

typedef __attribute__((ext_vector_type(2))) float v2f;
typedef __attribute__((ext_vector_type(8))) float v8f;

#define NB 32768
#define NF 1024
#define NL 4

// ---------------------------------------------------------------------------
// Kernel 1: e[i] = dot(c_i, W_i), csum = b0+b1+b2+b3.  One block, 1024 thr.
// ---------------------------------------------------------------------------
__global__ __launch_bounds__(1024) void cross_prep(
    const float* __restrict__ Ws, const float* __restrict__ Bs,
    float* __restrict__ e, float* __restrict__ csum) {
  __shared__ float s1[1024], s2[1024], s3[1024];
  const int f = threadIdx.x;

  const float b0 = Bs[0 * NF + f];
  const float b1 = Bs[1 * NF + f];
  const float b2 = Bs[2 * NF + f];
  const float b3 = Bs[3 * NF + f];
  const float c1 = b0;
  const float c2 = c1 + b1;
  const float c3 = c2 + b2;
  csum[f] = c3 + b3;

  s1[f] = c1 * Ws[1 * NF + f];
  s2[f] = c2 * Ws[2 * NF + f];
  s3[f] = c3 * Ws[3 * NF + f];
  __syncthreads();
  for (int stride = 512; stride > 0; stride >>= 1) {
    if (f < stride) {
      s1[f] += s1[f + stride];
      s2[f] += s2[f + stride];
      s3[f] += s3[f + stride];
    }
    __syncthreads();
  }
  if (f == 0) {
    e[0] = 0.0f;   // c_0 = 0
    e[1] = s1[0];
    e[2] = s2[0];
    e[3] = s3[0];
  }
}

// ---------------------------------------------------------------------------
// Kernel 2: d[row][l] = dot(x0[row,:], Ws[l,:])  via V_WMMA_F32_16X16X4_F32.
//
// Per wave: one 16-row tile.  A = x tile (16x4, fp32):
//   lane = 16*kh + r  holds A[M=r][K=2kh] , A[M=r][K=2kh+1]  (ISA 7.12.2)
// B = W^T (4x16), layer replicated over N via r&3:
//   lane holds B[K=2kh][N=r]=Ws[r&3][f+2kh], B[K=2kh+1][N=r]
// D (16x16 f32, 8 VGPRs): acc[g] = D[M = g + 8*kh][N = r].
// 128 threads/block = 4 waves = 64 rows/block.
// ---------------------------------------------------------------------------
__global__ __launch_bounds__(128) void cross_dots(
    const float* __restrict__ x, const float* __restrict__ Ws,
    float* __restrict__ d) {
  const int lane = threadIdx.x & 31;
  const int wave = threadIdx.x >> 5;
  const int r  = lane & 15;
  const int kh = lane >> 4;  // which half of the wave
  const int row_base = (blockIdx.x * 4 + wave) * 16;

  const float* __restrict__ xrow = x  + (size_t)(row_base + r) * NF + 2 * kh;
  const float* __restrict__ wrow = Ws + (size_t)(r & 3)        * NF + 2 * kh;

  v8f acc0 = {};
  v8f acc1 = {};
#pragma unroll 4
  for (int f = 0; f < NF; f += 8) {
    v2f a0 = *(const v2f*)(xrow + f);
    v2f b0 = *(const v2f*)(wrow + f);
    v2f a1 = *(const v2f*)(xrow + f + 4);
    v2f b1 = *(const v2f*)(wrow + f + 4);
    acc0 = __builtin_amdgcn_wmma_f32_16x16x4_f32(
        false, a0, false, b0, (short)0, acc0, false, false);
    acc1 = __builtin_amdgcn_wmma_f32_16x16x4_f32(
        false, a1, false, b1, (short)0, acc1, false, false);
  }

  if (r < NL) {  // lanes 0..3 (rows 0..7) and 16..19 (rows 8..15)
#pragma unroll
    for (int g = 0; g < 8; ++g) {
      d[(size_t)(row_base + g + 8 * kh) * NL + r] = acc0[g] + acc1[g];
    }
  }
}

// ---------------------------------------------------------------------------
// Kernel 3: a4[row] from the scalar recurrence.
// ---------------------------------------------------------------------------
__global__ __launch_bounds__(256) void cross_a4(
    const float* __restrict__ d, const float* __restrict__ e,
    float* __restrict__ a4) {
  const int row = blockIdx.x * 256 + threadIdx.x;
  if (row < NB) {
    float a = 1.0f;
#pragma unroll
    for (int i = 0; i < NL; ++i) {
      const float s = a * d[row * NL + i] + e[i];
      a += s;
    }
    a4[row] = a;
  }
}

// ---------------------------------------------------------------------------
// Kernel 4: out = a4[row] * x + csum[col].  float4 coalesced, 1 vec/thread.
// x should still be L2-resident (128 MB < 192 MB L2) from kernel 2's pass.
// ---------------------------------------------------------------------------
__global__ __launch_bounds__(256) void cross_finalize(
    const float* __restrict__ x, const float* __restrict__ a4,
    const float* __restrict__ csum, float* __restrict__ out) {
  const size_t idx  = (size_t)blockIdx.x * 256 + threadIdx.x;  // float4 index
  const size_t row  = idx >> 8;        // NF/4 = 256 float4 per row
  const size_t col4 = idx & 255;

  const float a = a4[row];
  const float4 xv = ((const float4*)x)[idx];
  const float4 cv = ((const float4*)csum)[col4];
  float4 o;
  o.x = fmaf(a, xv.x, cv.x);
  o.y = fmaf(a, xv.y, cv.y);
  o.z = fmaf(a, xv.z, cv.z);
  o.w = fmaf(a, xv.w, cv.w);
  ((float4*)out)[idx] = o;
}

// ---------------------------------------------------------------------------
extern "C" void kernel_launch(void* const* d_in, const int* in_sizes, int n_in,
                              void* d_out, int out_size, void* d_ws,
                              size_t ws_size, hipStream_t stream) {
  const float* x  = (const float*)d_in[0];   // [32768,1024]
  const float* Ws = (const float*)d_in[1];   // [4,1024]
  const float* Bs = (const float*)d_in[2];   // [4,1024]
  float* out = (float*)d_out;                // [32768,1024]

  // Scratch layout (floats): e[4] | csum[1024] | a4[32768] | d[32768*4]
  float* ws   = (float*)d_ws;
  float* e    = ws;                  // 4
  float* csum = ws + 4;              // 1024   (16B aligned)
  float* a4   = ws + 4 + 1024;       // 32768  (16B aligned)
  float* dmat = ws + 4 + 1024 + NB;  // 131072 (16B aligned)

  cross_prep<<<1, 1024, 0, stream>>>(Ws, Bs, e, csum);
  cross_dots<<<NB / 64, 128, 0, stream>>>(x, Ws, dmat);
  cross_a4<<<NB / 256, 256, 0, stream>>>(dmat, e, a4);
  cross_finalize<<<(NB * (NF / 4)) / 256, 256, 0, stream>>>(x, a4, csum, out);
}